// MockSIEVEModel_58798102282743
// MI455X (gfx1250) — compile-verified
//
#include <hip/hip_runtime.h>
#include <hip/hip_bf16.h>

// Problem constants (from reference setup_inputs)
#define BB   32
#define VV   4096
#define DD   64
#define GG   20000

typedef __attribute__((ext_vector_type(2))) float v2f;
typedef __attribute__((ext_vector_type(8))) float v8f;

// ---------------------------------------------------------------------------
// Pass 0: init winner table to -1, init out[b] = bias
// ---------------------------------------------------------------------------
__global__ void sieve_init_kernel(int* __restrict__ winner,
                                  float* __restrict__ out,
                                  const float* __restrict__ bias) {
    int i = blockIdx.x * blockDim.x + threadIdx.x;
    if (i < BB * GG) winner[i] = -1;
    if (i < BB) out[i] = bias[0];
}

// ---------------------------------------------------------------------------
// Pass 1: last-write-wins masked scatter  ->  winner[b][g] = max valid v
// ---------------------------------------------------------------------------
__global__ void sieve_scatter_kernel(const int* __restrict__ gene_ids,
                                     const int* __restrict__ mask,
                                     int* __restrict__ winner) {
    int i = blockIdx.x * blockDim.x + threadIdx.x;
    if (i >= BB * VV) return;
    int b = i / VV;
    int v = i - b * VV;
    int gid = gene_ids[i];
    if (mask[i] > 0 && gid >= 0 && gid < GG) {
        atomicMax(&winner[b * GG + gid], v);
    }
}

// ---------------------------------------------------------------------------
// Pass 2: per 16-token tile, D = A(16x64) x Bw(64x16) via 16x V_WMMA_F32_16X16X4_F32.
//   A row m  = winner-token feature row (zeroed if token not a winner)
//   Bw col n = W row of token n's gene (gathered; W is L2-resident, 5 MB)
//   diag(D)  = per-token dot products -> wave reduce -> atomicAdd out[b]
// One wave (32 lanes) per tile; 8192 tiles total. EXEC stays all-1s.
// ---------------------------------------------------------------------------
__global__ void sieve_wmma_kernel(const float* __restrict__ feat,
                                  const int*   __restrict__ gene_ids,
                                  const int*   __restrict__ mask,
                                  const float* __restrict__ W,
                                  const int*   __restrict__ winner,
                                  float* __restrict__ out) {
    const int lane = threadIdx.x & 31;
    const int wave = threadIdx.x >> 5;
    const int tile = blockIdx.x * (blockDim.x >> 5) + wave;   // 0 .. 8191

    const int tilesPerBatch = VV / 16;                         // 256
    const int b  = tile / tilesPerBatch;
    const int v0 = (tile - b * tilesPerBatch) * 16;

    // WMMA f32 16x16x4 A-layout: lanes 0-15 hold K=k0,k0+1 ; lanes 16-31 hold K=k0+2,k0+3
    const int m  = lane & 15;         // token index within tile / matrix row M
    const int hi = lane >> 4;         // K half selector
    const int t  = v0 + m;            // token id in [0, V)

    const long tokIdx = (long)b * VV + t;
    int gid = gene_ids[tokIdx];
    int msk = mask[tokIdx];
    bool valid = (msk > 0) && (gid >= 0) && (gid < GG);
    if (valid) valid = (winner[b * GG + gid] == t);            // last-write-wins winner?
    const int wrow = valid ? gid : 0;                          // safe gather row
    const float scale = valid ? 1.0f : 0.0f;                   // zero A rows of losers

    const float* __restrict__ frow = feat + tokIdx * DD;       // token feature row
    const float* __restrict__ wr   = W + (long)wrow * DD;      // gathered W row

    v8f c = {};                                                // 16x16 f32 accumulator
#pragma unroll
    for (int k0 = 0; k0 < DD; k0 += 4) {
        const int k = k0 + 2 * hi;
        const float2 fa = *(const float2*)(frow + k);          // global_load_b64
        const float2 wb = *(const float2*)(wr + k);            // global_load_b64
        v2f a;  a.x = fa.x * scale;  a.y = fa.y * scale;
        v2f bm; bm.x = wb.x;         bm.y = wb.y;
        // D = A x B + C   (exact f32 WMMA)
        c = __builtin_amdgcn_wmma_f32_16x16x4_f32(
                /*neg_a=*/false, a, /*neg_b=*/false, bm,
                /*c_mod=*/(short)0, c, /*reuse_a=*/false, /*reuse_b=*/false);
    }

    // Extract diag(D): (i,i) at {lane i, vgpr i} for i<8 ; {lane 24+i', vgpr i'} for i>=8
    float dval = 0.0f;
#pragma unroll
    for (int r = 0; r < 8; ++r) {
        dval = (lane == r)      ? c[r] : dval;
        dval = (lane == 24 + r) ? c[r] : dval;
    }

    // wave32 reduction
#pragma unroll
    for (int off = 16; off > 0; off >>= 1)
        dval += __shfl_xor(dval, off, 32);

    if (lane == 0) atomicAdd(&out[b], dval);
}

// ---------------------------------------------------------------------------
extern "C" void kernel_launch(void* const* d_in, const int* in_sizes, int n_in,
                              void* d_out, int out_size, void* d_ws, size_t ws_size,
                              hipStream_t stream) {
    const float* features = (const float*)d_in[0];   // (B, V, D) f32
    // d_in[1] = positions (unused by reference math)
    const int*   gene_ids = (const int*)d_in[2];     // (B, V) i32
    const int*   mask     = (const int*)d_in[3];     // (B, V) i32
    const float* W        = (const float*)d_in[4];   // (1, G*D) f32
    const float* bias     = (const float*)d_in[5];   // (1,) f32

    float* out   = (float*)d_out;                    // (B, 1) f32
    int* winner  = (int*)d_ws;                       // B*G ints = 2.56 MB scratch

    // Pass 0: init (covers B*G winner entries + B outputs)
    {
        int n = BB * GG;
        sieve_init_kernel<<<(n + 255) / 256, 256, 0, stream>>>(winner, out, bias);
    }
    // Pass 1: scatter-max
    {
        int n = BB * VV;
        sieve_scatter_kernel<<<(n + 255) / 256, 256, 0, stream>>>(gene_ids, mask, winner);
    }
    // Pass 2: WMMA dot-accumulate. 8192 tiles, 8 waves (256 thr) per block -> 1024 blocks.
    {
        int tiles = BB * (VV / 16);
        int wavesPerBlock = 8;
        int blocks = tiles / wavesPerBlock;
        sieve_wmma_kernel<<<blocks, wavesPerBlock * 32, 0, stream>>>(
            features, gene_ids, mask, W, winner, out);
    }
}